// TimeConsistencyLoss_80367428043216
// MI455X (gfx1250) — compile-verified
//
#include <hip/hip_runtime.h>

// Problem constants (match reference setup_inputs)
#define B_   16
#define T_   32641
#define E_   128
#define L_   32768          // T + E - 1
#define TILE 256            // rows per k1 block
#define NT   128            // ceil(T / TILE)
#define K1_BLOCKS (B_ * NT) // 2048
#define K2_BLOCKS 512
#define DP_STRIDE 384       // TILE + E - 1 rounded up (383 -> 384)

typedef float v2f __attribute__((ext_vector_type(2)));
typedef float v8f __attribute__((ext_vector_type(8)));

// Full-f32 wave reduction using V_WMMA_F32_16X16X4_F32:
// A = ones(16x4), B carries one value per lane (64 B-slots, half zero).
// C[m][n] = sum_k B[k][n]; all C rows identical, so summing lane-held
// column values over a 16-lane butterfly yields the total in every lane.
// Layout-agnostic: we only need the SUM of all B elements.
__device__ __forceinline__ float wave_sum_wmma(float x) {
  v2f a; a[0] = 1.0f; a[1] = 1.0f;
  v2f b; b[0] = x;    b[1] = 0.0f;
  v8f c = {};
  c = __builtin_amdgcn_wmma_f32_16x16x4_f32(false, a, false, b, (short)0, c,
                                            false, false);
  float s = c[0];                 // lane n holds C[0][n] (lanes 16..31 mirror)
  s += __shfl_xor(s, 1, 32);
  s += __shfl_xor(s, 2, 32);
  s += __shfl_xor(s, 4, 32);
  s += __shfl_xor(s, 8, 32);
  return s;                       // total in every lane
}

// blockDim.x must be 256 (8 waves). Result valid on thread 0.
__device__ __forceinline__ float block_sum(float x, float* red8) {
  const float ws = wave_sum_wmma(x);
  const int wave = threadIdx.x >> 5;
  const int lane = threadIdx.x & 31;
  if (lane == 0) red8[wave] = ws;
  __syncthreads();
  float tot = 0.0f;
  if (threadIdx.x == 0) {
#pragma unroll
    for (int w = 0; w < 8; ++w) tot += red8[w];
  }
  return tot;
}

// count of elements in group g: c = min(g, T-1) - max(0, g-(E-1)) + 1
__device__ __forceinline__ float group_inv_count(int g) {
  const int lo = g > (E_ - 1) ? g - (E_ - 1) : 0;
  const int hi = g < T_ ? g : (T_ - 1);
  return 1.0f / (float)(hi - lo + 1);
}

// Kernel 1: stream one (batch, row-tile); build diagonal partial sums in LDS
// (ds_add_f32 scatter), emit 384 partials + weighted sum-of-squares partial.
__global__ __launch_bounds__(256) void tc_k1(const float* __restrict__ x,
                                             float* __restrict__ dp,
                                             float* __restrict__ part1) {
  __shared__ float diag[DP_STRIDE];
  __shared__ float red8[8];
  const int tid  = threadIdx.x;
  const int tile = blockIdx.x;
  const int b    = blockIdx.y;

  for (int k = tid; k < DP_STRIDE; k += 256) diag[k] = 0.0f;
  __syncthreads();

  const int j0 = tile * TILE;
  const float* xb = x + (size_t)b * ((size_t)T_ * E_);
  const int wave = tid >> 5;
  const int lane = tid & 31;
  // whole tile lies in groups [E-1, T-1] -> count == 128 everywhere
  const bool interior = (j0 >= E_ - 1) && (j0 + TILE - 1 + E_ - 1 <= T_ - 1);

  float t1 = 0.0f;
  for (int r = wave; r < TILE; r += 8) {
    const int j = j0 + r;                 // uniform across the wave
    if (j >= T_) break;
    const float4 v = *(const float4*)(xb + (size_t)j * E_ + 4 * lane);
    const int base = r + 4 * lane;        // local diagonal slot
    atomicAdd(&diag[base + 0], v.x);      // ds_add_f32, distinct banks/lane
    atomicAdd(&diag[base + 1], v.y);
    atomicAdd(&diag[base + 2], v.z);
    atomicAdd(&diag[base + 3], v.w);
    if (interior) {
      t1 += v.x * v.x + v.y * v.y + v.z * v.z + v.w * v.w;  // * 1/128 later
    } else {
      const int g0 = j + 4 * lane;
      const float e[4] = {v.x, v.y, v.z, v.w};
#pragma unroll
      for (int q = 0; q < 4; ++q) {
        const float w = group_inv_count(g0 + q);
        t1 += e[q] * e[q] * w;
      }
    }
  }
  if (interior) t1 *= (1.0f / 128.0f);

  __syncthreads();
  float* dpt = dp + (size_t)(b * NT + tile) * DP_STRIDE;
  for (int k = tid; k < DP_STRIDE; k += 256) dpt[k] = diag[k];

  const float bt = block_sum(t1, red8);
  if (tid == 0) part1[b * NT + tile] = bt;
}

// Kernel 2: S1(b,t) = tile partial (t>>8) + trailing partial of previous tile;
// accumulate S1^2 * w^2 and reduce.
__global__ __launch_bounds__(256) void tc_k2(const float* __restrict__ dp,
                                             float* __restrict__ part2) {
  __shared__ float red8[8];
  const int tid = threadIdx.x;
  float acc = 0.0f;
#pragma unroll
  for (int k = 0; k < 4; ++k) {
    const int idx = (blockIdx.x * 256 + tid) + k * (K2_BLOCKS * 256);
    const int b  = idx >> 15;             // L = 32768
    const int t  = idx & (L_ - 1);
    const int i0 = t >> 8;                // owning tile
    const int r  = t & 255;
    float s = dp[(size_t)(b * NT + i0) * DP_STRIDE + r];
    if (r <= E_ - 2 && i0 > 0)            // overlap from previous tile
      s += dp[(size_t)(b * NT + i0 - 1) * DP_STRIDE + (r + 256)];
    const float w = group_inv_count(t);
    acc += s * s * w * w;
  }
  const float bt = block_sum(acc, red8);
  if (tid == 0) part2[blockIdx.x] = bt;
}

// Kernel 3: deterministic final reduction and scale.
__global__ __launch_bounds__(256) void tc_k3(const float* __restrict__ part1,
                                             const float* __restrict__ part2,
                                             float* __restrict__ out) {
  __shared__ float sm[256];
  const int tid = threadIdx.x;
  float a = 0.0f;
  for (int k = tid; k < K1_BLOCKS; k += 256) a += part1[k];
  for (int k = tid; k < K2_BLOCKS; k += 256) a -= part2[k];
  sm[tid] = a;
  __syncthreads();
  for (int s = 128; s > 0; s >>= 1) {
    if (tid < s) sm[tid] += sm[tid + s];
    __syncthreads();
  }
  if (tid == 0) out[0] = sm[0] * (0.1f / 524288.0f);  // 0.1 / (B*L)
}

extern "C" void kernel_launch(void* const* d_in, const int* in_sizes, int n_in,
                              void* d_out, int out_size, void* d_ws,
                              size_t ws_size, hipStream_t stream) {
  (void)in_sizes; (void)n_in; (void)out_size; (void)ws_size;
  const float* x = (const float*)d_in[0];

  float* dp    = (float*)d_ws;                          // 16*128*384 floats
  float* part1 = dp + (size_t)B_ * NT * DP_STRIDE;      // 2048 floats
  float* part2 = part1 + K1_BLOCKS;                     // 512 floats

  dim3 g1(NT, B_);
  tc_k1<<<g1, 256, 0, stream>>>(x, dp, part1);
  tc_k2<<<K2_BLOCKS, 256, 0, stream>>>(dp, part2);
  tc_k3<<<1, 256, 0, stream>>>(part1, part2, (float*)d_out);
}